// SimpleGCN_16054587752866
// MI455X (gfx1250) — compile-verified
//
#include <hip/hip_runtime.h>
#include <hip/hip_bf16.h>
#include <math.h>

// ---------------------------------------------------------------------------
// GIN forward for MI455X (gfx1250, wave32).
// Dense layers use V_WMMA_F32_16X16X4_F32 (fp32-exact matrix path), persistent
// blocks stage the 64KB weight matrix into LDS once and grid-stride row tiles.
// Scatter/pool use global_atomic_add_f32 (working set is L2-resident: 192MB L2).
// ---------------------------------------------------------------------------

typedef __attribute__((ext_vector_type(2))) float v2f;
typedef __attribute__((ext_vector_type(8))) float v8f;

#define DFEAT 132           // padded LDS row stride (128 + 4) -> bank-spread, float2/float4 aligned

// ---------------------------------------------------------------------------
// zero fill
__global__ __launch_bounds__(256) void zero_kernel(float* __restrict__ p, long n) {
    long t = (long)blockIdx.x * 256 + threadIdx.x;
    if (t < n) p[t] = 0.0f;
}

// ---------------------------------------------------------------------------
// agg[dst] += x[src]  (edge-parallel; 32 lanes per edge, 4 feats per lane)
__global__ __launch_bounds__(256) void scatter_add_kernel(
    const float* __restrict__ X, const int* __restrict__ ei,
    float* __restrict__ AGG, int n_edges) {
    long t = (long)blockIdx.x * 256 + threadIdx.x;
    int e = (int)(t >> 5);
    if (e >= n_edges) return;
    int c = ((int)t & 31) * 4;
    int src = ei[e];
    int dst = ei[n_edges + e];
    float4 v = *(const float4*)(X + (size_t)src * 128 + c);
    float* p = AGG + (size_t)dst * 128 + c;
    atomicAdd(p + 0, v.x);
    atomicAdd(p + 1, v.y);
    atomicAdd(p + 2, v.z);
    atomicAdd(p + 3, v.w);
}

// ---------------------------------------------------------------------------
// OUT = act( (X (+AGG)) @ W + bias ), W is [128 in][128 out] row-major.
// Persistent: block = 256 thr (8 waves); W staged (transposed) into LDS once,
// then grid-stride over 16-row tiles. Wave w owns output cols [16w,16w+16).
// 32 x v_wmma_f32_16x16x4_f32 per wave per tile, K fully unrolled.
__global__ __launch_bounds__(256) void gin_gemm_kernel(
    const float* __restrict__ X, const float* __restrict__ AGG,
    const float* __restrict__ W, const float* __restrict__ bias,
    float* __restrict__ OUT, int n_rows, int n_tiles, int relu) {
    extern __shared__ float lds[];
    float* Wt = lds;                 // [128 n][DFEAT] : Wt[n*DFEAT + k] = W[k][n]
    float* As = lds + 128 * DFEAT;   // [16 m][DFEAT]

    const int tid = threadIdx.x;

    // Stage W transposed into LDS, ONCE (coalesced global reads, scalar LDS stores).
    for (int i = tid; i < 128 * 32; i += 256) {
        int k = i >> 5;
        int n = (i & 31) * 4;
        float4 w4 = *(const float4*)(W + (size_t)k * 128 + n);
        Wt[(n + 0) * DFEAT + k] = w4.x;
        Wt[(n + 1) * DFEAT + k] = w4.y;
        Wt[(n + 2) * DFEAT + k] = w4.z;
        Wt[(n + 3) * DFEAT + k] = w4.w;
    }

    const int wave = tid >> 5;
    const int lane = tid & 31;
    const int nl   = lane & 15;          // M (for A) / N (for B) within tile
    const int koff = (lane >> 4) * 2;    // lanes 16..31 carry K+2,K+3
    const int n0   = wave * 16;
    const int mhi  = (lane >> 4) * 8;    // C layout: VGPR r -> M=r (lanes 0-15), M=r+8 (16-31)
    const int Nc   = n0 + nl;
    const float bv = bias[Nc];

    const float* arow = As + nl * DFEAT;
    const float* brow = Wt + (n0 + nl) * DFEAT;

    for (int tile = blockIdx.x; tile < n_tiles; tile += gridDim.x) {
        const int row0 = tile * 16;

        __syncthreads();   // prior-iter LDS reads done (also orders W staging on iter 0)
        // Stage A tile (16 x 128), fusing the GIN "x + agg" add.
        for (int i = tid; i < 16 * 32; i += 256) {
            int r = i >> 5;
            int c = (i & 31) * 4;
            float4 v = make_float4(0.f, 0.f, 0.f, 0.f);
            int row = row0 + r;
            if (row < n_rows) {
                size_t g = (size_t)row * 128 + c;
                v = *(const float4*)(X + g);
                if (AGG) {
                    float4 a = *(const float4*)(AGG + g);
                    v.x += a.x; v.y += a.y; v.z += a.z; v.w += a.w;
                }
            }
            *(float4*)(As + r * DFEAT + c) = v;
        }
        __syncthreads();

        v8f acc = {};
#pragma unroll
        for (int k0 = 0; k0 < 128; k0 += 4) {
            v2f a = *(const v2f*)(arow + k0 + koff);
            v2f b = *(const v2f*)(brow + k0 + koff);
            acc = __builtin_amdgcn_wmma_f32_16x16x4_f32(
                false, a, false, b, (short)0, acc, false, false);
        }

#pragma unroll
        for (int r = 0; r < 8; ++r) {
            int row = row0 + r + mhi;
            if (row < n_rows) {
                float v = acc[r] + bv;
                if (relu) v = fmaxf(v, 0.0f);
                OUT[(size_t)row * 128 + Nc] = v;
            }
        }
    }
}

// ---------------------------------------------------------------------------
// Per-feature sum / sumsq partials (block = 128 thr = 1 feature each, 256-row chunk).
__global__ __launch_bounds__(128) void bn_stats_kernel(
    const float* __restrict__ H, float* __restrict__ stats, int n_rows) {
    int f  = threadIdx.x;
    int r0 = blockIdx.x * 256;
    int r1 = r0 + 256; if (r1 > n_rows) r1 = n_rows;
    float s = 0.f, s2 = 0.f;
    for (int r = r0; r < r1; ++r) {
        float v = H[(size_t)r * 128 + f];
        s += v; s2 += v * v;
    }
    atomicAdd(&stats[f], s);
    atomicAdd(&stats[128 + f], s2);
}

// stats -> per-feature scale/shift
__global__ __launch_bounds__(128) void bn_finalize_kernel(
    const float* __restrict__ stats, const float* __restrict__ gamma,
    const float* __restrict__ beta, float* __restrict__ ss, float inv_n) {
    int f = threadIdx.x;
    float mean = stats[f] * inv_n;
    float var  = stats[128 + f] * inv_n - mean * mean;
    float sc   = gamma[f] * rsqrtf(var + 1e-5f);
    ss[f]       = sc;
    ss[128 + f] = beta[f] - mean * sc;
}

// y = max(scale*x + shift, 0) in place, float4 per thread
__global__ __launch_bounds__(256) void bn_apply_relu_kernel(
    float* __restrict__ H, const float* __restrict__ ss, long n_vec4) {
    long t = (long)blockIdx.x * 256 + threadIdx.x;
    if (t >= n_vec4) return;
    int c = ((int)t & 31) * 4;
    float4 v = *(float4*)(H + t * 4);
    v.x = fmaxf(v.x * ss[c + 0] + ss[128 + c + 0], 0.f);
    v.y = fmaxf(v.y * ss[c + 1] + ss[128 + c + 1], 0.f);
    v.z = fmaxf(v.z * ss[c + 2] + ss[128 + c + 2], 0.f);
    v.w = fmaxf(v.w * ss[c + 3] + ss[128 + c + 3], 0.f);
    *(float4*)(H + t * 4) = v;
}

// ---------------------------------------------------------------------------
// pooled[batch[n]] += h[n];  cnt[batch[n]] += 1  (32 lanes per node)
__global__ __launch_bounds__(256) void pool_kernel(
    const float* __restrict__ H, const int* __restrict__ batch,
    float* __restrict__ pooled, float* __restrict__ cnt, int n_rows) {
    long t = (long)blockIdx.x * 256 + threadIdx.x;
    int n = (int)(t >> 5);
    if (n >= n_rows) return;
    int c = ((int)t & 31) * 4;
    int g = batch[n];
    float4 v = *(const float4*)(H + (size_t)n * 128 + c);
    float* p = pooled + (size_t)g * 128 + c;
    atomicAdd(p + 0, v.x);
    atomicAdd(p + 1, v.y);
    atomicAdd(p + 2, v.z);
    atomicAdd(p + 3, v.w);
    if (((int)t & 31) == 0) atomicAdd(&cnt[g], 1.0f);
}

// ---------------------------------------------------------------------------
// logits = (pooled/cnt) @ wl + bl ; log_softmax.  One block, thread = graph.
__global__ __launch_bounds__(256) void head_kernel(
    const float* __restrict__ pooled, const float* __restrict__ cnt,
    const float* __restrict__ wl, const float* __restrict__ bl,
    float* __restrict__ out) {
    int g = threadIdx.x;  // 0..255
    float inv = 1.0f / fmaxf(cnt[g], 1.0f);
    float lg[7];
#pragma unroll
    for (int j = 0; j < 7; ++j) lg[j] = bl[j];
    for (int k = 0; k < 128; ++k) {
        float p = pooled[(size_t)g * 128 + k] * inv;
#pragma unroll
        for (int j = 0; j < 7; ++j) lg[j] += p * wl[k * 7 + j];
    }
    float mx = lg[0];
#pragma unroll
    for (int j = 1; j < 7; ++j) mx = fmaxf(mx, lg[j]);
    float se = 0.f;
#pragma unroll
    for (int j = 0; j < 7; ++j) se += expf(lg[j] - mx);
    float lse = mx + logf(se);
#pragma unroll
    for (int j = 0; j < 7; ++j) out[g * 7 + j] = lg[j] - lse;
}

// ---------------------------------------------------------------------------
extern "C" void kernel_launch(void* const* d_in, const int* in_sizes, int n_in,
                              void* d_out, int out_size, void* d_ws, size_t ws_size,
                              hipStream_t stream) {
    const float* x   = (const float*)d_in[0];
    const float* w1a = (const float*)d_in[1];
    const float* b1a = (const float*)d_in[2];
    const float* w1b = (const float*)d_in[3];
    const float* b1b = (const float*)d_in[4];
    const float* g1  = (const float*)d_in[5];
    const float* be1 = (const float*)d_in[6];
    const float* w2a = (const float*)d_in[7];
    const float* b2a = (const float*)d_in[8];
    const float* w2b = (const float*)d_in[9];
    const float* b2b = (const float*)d_in[10];
    const float* g2  = (const float*)d_in[11];
    const float* be2 = (const float*)d_in[12];
    const float* wl  = (const float*)d_in[13];
    const float* bl  = (const float*)d_in[14];
    const int*   ei  = (const int*)d_in[15];
    const int*   bat = (const int*)d_in[16];
    float* out = (float*)d_out;

    const int NN = in_sizes[0] / 128;       // 100000
    const int NE = in_sizes[15] / 2;        // 1600000
    const long NF = (long)NN * 128;         // 12.8M floats

    // workspace layout (floats): B1 | B2 | B3 | stats(256) | ss(256) | pooled(256*128) | cnt(256)
    float* B1     = (float*)d_ws;
    float* B2     = B1 + NF;
    float* B3     = B2 + NF;
    float* stats  = B3 + NF;
    float* ss     = stats + 256;
    float* pooled = ss + 256;
    float* cnt    = pooled + 256 * 128;

    const int zgrid   = (int)((NF + 255) / 256);
    const int sgrid   = (int)(((long)NE * 32 + 255) / 256);
    const int ntiles  = (NN + 15) / 16;                   // 6250
    int ggrid = ntiles < 2048 ? ntiles : 2048;            // persistent blocks
    const int bgrid   = (NN + 255) / 256;
    const int egrid   = (int)((NF / 4 + 255) / 256);
    const int pgrid   = (int)(((long)NN * 32 + 255) / 256);
    const size_t gemm_lds = (size_t)(128 * DFEAT + 16 * DFEAT) * sizeof(float); // ~76 KB (320 KB WGP budget)
    const float inv_n = 1.0f / (float)NN;

    // ---- conv1: agg = scatter(x); h = relu((x+agg)W1a+b1a) @ W1b + b1b
    zero_kernel<<<zgrid, 256, 0, stream>>>(B1, NF);
    scatter_add_kernel<<<sgrid, 256, 0, stream>>>(x, ei, B1, NE);
    gin_gemm_kernel<<<ggrid, 256, gemm_lds, stream>>>(x, B1, w1a, b1a, B2, NN, ntiles, 1);
    gin_gemm_kernel<<<ggrid, 256, gemm_lds, stream>>>(B2, nullptr, w1b, b1b, B3, NN, ntiles, 0);

    // ---- BN1 + ReLU (training stats, biased var)
    zero_kernel<<<1, 256, 0, stream>>>(stats, 256);
    bn_stats_kernel<<<bgrid, 128, 0, stream>>>(B3, stats, NN);
    bn_finalize_kernel<<<1, 128, 0, stream>>>(stats, g1, be1, ss, inv_n);
    bn_apply_relu_kernel<<<egrid, 256, 0, stream>>>(B3, ss, NF / 4);

    // ---- conv2
    zero_kernel<<<zgrid, 256, 0, stream>>>(B1, NF);
    scatter_add_kernel<<<sgrid, 256, 0, stream>>>(B3, ei, B1, NE);
    gin_gemm_kernel<<<ggrid, 256, gemm_lds, stream>>>(B3, B1, w2a, b2a, B2, NN, ntiles, 1);
    gin_gemm_kernel<<<ggrid, 256, gemm_lds, stream>>>(B2, nullptr, w2b, b2b, B1, NN, ntiles, 0);

    // ---- BN2 + ReLU
    zero_kernel<<<1, 256, 0, stream>>>(stats, 256);
    bn_stats_kernel<<<bgrid, 128, 0, stream>>>(B1, stats, NN);
    bn_finalize_kernel<<<1, 128, 0, stream>>>(stats, g2, be2, ss, inv_n);
    bn_apply_relu_kernel<<<egrid, 256, 0, stream>>>(B1, ss, NF / 4);

    // ---- global mean pool + head + log_softmax
    zero_kernel<<<(256 * 128 + 256 + 255) / 256 + 1, 256, 0, stream>>>(pooled, 256 * 128 + 256);
    pool_kernel<<<pgrid, 256, 0, stream>>>(B1, bat, pooled, cnt, NN);
    head_kernel<<<1, 256, 0, stream>>>(pooled, cnt, wl, bl, out);

    (void)n_in; (void)out_size; (void)ws_size;
}